// RNNModelScratch_53987738911265
// MI455X (gfx1250) — compile-verified
//
#include <hip/hip_runtime.h>
#include <hip/hip_bf16.h>
#include <math.h>

#define VOCAB 10000
#define HID   256
#define BATCH 32
#define STEPS 256

typedef __attribute__((ext_vector_type(2))) float v2f;
typedef __attribute__((ext_vector_type(8))) float v8f;

// D = A(16x4 f32) x B(4x16 f32) + C(16x16 f32), full f32 precision (matches reference).
// A layout: lanes 0-15 hold M=lane, K={0,1} in a.x/a.y; lanes 16-31 hold M=lane-16, K={2,3}.
// B layout: lanes 0-15 hold N=lane, K={0,1} in b.x/b.y; lanes 16-31 hold N=lane-16, K={2,3}.
// C/D layout: VGPR v, lanes 0-15 -> (M=v, N=lane); lanes 16-31 -> (M=8+v, N=lane-16).
__device__ __forceinline__ v8f wmma_f32_4(v2f a, v2f b, v8f c) {
  return __builtin_amdgcn_wmma_f32_16x16x4_f32(
      /*neg_a=*/false, a, /*neg_b=*/false, b,
      /*c_mod=*/(short)0, c, /*reuse_a=*/false, /*reuse_b=*/false);
}

// CDNA5 async DMA: copy 16 bytes per lane global -> LDS, tracked by ASYNCcnt.
// VDST holds the 32-bit LDS byte offset (low 32 bits of the generic shared
// address).
__device__ __forceinline__ void async_g2lds_b128(void* lds, const void* gaddr) {
  unsigned ldsOff = (unsigned)(size_t)lds;
  asm volatile("global_load_async_to_lds_b128 %0, %1, off"
               :: "v"(ldsOff), "v"(gaddr) : "memory");
}
__device__ __forceinline__ void wait_asynccnt0() {
#if __has_builtin(__builtin_amdgcn_s_wait_asynccnt)
  __builtin_amdgcn_s_wait_asynccnt(0);
#else
  asm volatile("s_wait_asynccnt 0x0" ::: "memory");
#endif
}

// ---------------------------------------------------------------------------
// Phase 1: sequential GRU recurrence. One workgroup, 32 waves.
// Wave w owns output tile (m = w>>4 in {0,1} batch-tiles, n = w&15 col-tiles).
// h and r*h are kept TRANSPOSED in LDS: hT[k*32 + b], so A-fragment loads
// (varying M across lanes, fixed K) hit consecutive LDS addresses.
// Weights W_h* stream from L2 (768 KB/step). The one-hot matmul is a gather
// of row W_x*[X[b,t]].
// ---------------------------------------------------------------------------
__global__ __launch_bounds__(1024, 1)
void gru_recurrence(const int* __restrict__ X, const float* __restrict__ H0,
                    const float* __restrict__ Wxz, const float* __restrict__ Whz,
                    const float* __restrict__ bz,
                    const float* __restrict__ Wxr, const float* __restrict__ Whr,
                    const float* __restrict__ br,
                    const float* __restrict__ Wxh, const float* __restrict__ Whh,
                    const float* __restrict__ bh,
                    float* __restrict__ Hall, float* __restrict__ Hfinal) {
  __shared__ float hT [HID * BATCH];   // hT[k*32 + b]  (32 KB)
  __shared__ float rhT[HID * BATCH];   // (r*h)^T       (32 KB)

  const int tid  = threadIdx.x;
  const int wave = tid >> 5;
  const int lane = tid & 31;
  const int i    = lane & 15;
  const int half = lane >> 4;
  const int bm   = (wave >> 4) * 16;   // batch-tile base (0 or 16)
  const int cn   = (wave & 15) * 16;   // column-tile base (0..240)

  for (int idx = tid; idx < BATCH * HID; idx += 1024) {
    const int b = idx >> 8, k = idx & 255;
    hT[k * BATCH + b] = H0[idx];
  }
  __syncthreads();

  const int   c   = cn + i;            // this lane's column (fixed)
  const float bzc = bz[c], brc = br[c], bhc = bh[c];

  for (int t = 0; t < STEPS; ++t) {
    // --- z and r gates: share the A fragment between both WMMAs ---
    v8f accz = {}, accr = {};
    for (int k0 = 0; k0 < HID; k0 += 4) {
      const int ka = k0 + 2 * half;
      v2f a, wz, wr;
      a.x  = hT[(ka + 0) * BATCH + bm + i];
      a.y  = hT[(ka + 1) * BATCH + bm + i];
      wz.x = Whz[(ka + 0) * HID + c];
      wz.y = Whz[(ka + 1) * HID + c];
      wr.x = Whr[(ka + 0) * HID + c];
      wr.y = Whr[(ka + 1) * HID + c];
      accz = wmma_f32_4(a, wz, accz);
      accr = wmma_f32_4(a, wr, accr);
    }

    float zv[8], ho[8];
    int   tokc[8];
    for (int v = 0; v < 8; ++v) {
      const int b   = bm + half * 8 + v;
      const int tok = X[b * STEPS + t];
      tokc[v] = tok;
      const float sz = accz[v] + Wxz[tok * HID + c] + bzc;
      const float sr = accr[v] + Wxr[tok * HID + c] + brc;
      const float z  = 1.f / (1.f + __expf(-sz));
      const float r  = 1.f / (1.f + __expf(-sr));
      const float h0 = hT[c * BATCH + b];
      zv[v] = z; ho[v] = h0;
      rhT[c * BATCH + b] = r * h0;
    }
    __syncthreads();  // all of (r*h)^T visible

    // --- candidate gate: (r*h) @ W_hh ---
    v8f acch = {};
    for (int k0 = 0; k0 < HID; k0 += 4) {
      const int ka = k0 + 2 * half;
      v2f a, wh;
      a.x  = rhT[(ka + 0) * BATCH + bm + i];
      a.y  = rhT[(ka + 1) * BATCH + bm + i];
      wh.x = Whh[(ka + 0) * HID + c];
      wh.y = Whh[(ka + 1) * HID + c];
      acch = wmma_f32_4(a, wh, acch);
    }

    for (int v = 0; v < 8; ++v) {
      const int   b  = bm + half * 8 + v;
      const float sh = acch[v] + Wxh[tokc[v] * HID + c] + bhc;
      const float ht = tanhf(sh);
      const float hn = zv[v] * ho[v] + (1.f - zv[v]) * ht;
      hT[c * BATCH + b] = hn;                     // tiles are disjoint: no race
      Hall[(t * BATCH + b) * HID + c] = hn;       // stacked H for phase 2
      if (t == STEPS - 1) Hfinal[b * HID + c] = hn;
    }
    __syncthreads();  // h_new complete before next step's matmuls
  }
}

// ---------------------------------------------------------------------------
// Phase 2: out = Hall(8192x256) @ W_hq(256x10000) + b_q. Memory-bound on the
// 328 MB output; 128x128 block tiles keep L2 traffic ~1.3 GB. Staging uses
// GLOBAL_LOAD_ASYNC_TO_LDS_B128 (no VGPR round-trip; ASYNCcnt-tracked).
// As pitch 36 floats: banks i*36 mod 64 distinct over 16 rows, halves disjoint,
// 144-B rows keep B128 16-B alignment. Bs pitch 136: the +2-row offset of
// lanes 16-31 shifts banks by exactly 16 -> conflict-free 32-lane reads.
// ---------------------------------------------------------------------------
#define TM 128
#define TN 128
#define TK 32
#define APITCH (TK + 4)   // 36 floats = 144 B
#define BPITCH (TN + 8)   // 136 floats = 544 B

__global__ __launch_bounds__(256, 2)
void proj_gemm(const float* __restrict__ Hall, const float* __restrict__ Whq,
               const float* __restrict__ bq, float* __restrict__ out) {
  __shared__ float As[TM][APITCH];  // As[m][k], row-major (async-friendly)
  __shared__ float Bs[TK][BPITCH];  // Bs[k][n]

  const int tid  = threadIdx.x;
  const int lane = tid & 31;
  const int wave = tid >> 5;
  const int i    = lane & 15;
  const int half = lane >> 4;

  const int nBase = blockIdx.x * TN;
  const int mBase = blockIdx.y * TM;
  const int rowBase = (wave & 3) * 32;  // wave's rows within block
  const int colBase = (wave >> 2) * 64; // wave's cols within block
  const bool fullN  = (nBase + TN) <= VOCAB;

  v8f acc[2][4];
  for (int a0 = 0; a0 < 2; ++a0)
    for (int b0 = 0; b0 < 4; ++b0) {
      v8f z = {};
      acc[a0][b0] = z;
    }

  for (int kc = 0; kc < HID; kc += TK) {
    // Stage A: 128 rows x 32 k. One float4 per thread per rep, DMA'd to LDS.
    for (int rep = 0; rep < 4; ++rep) {
      const int idx = rep * 256 + tid;
      const int kk4 = (idx & 7) * 4;   // 0,4,...,28
      const int mm  = idx >> 3;        // 0..127
      async_g2lds_b128(&As[mm][kk4],
                       &Hall[(size_t)(mBase + mm) * HID + kc + kk4]);
    }
    // Stage B: 32 k-rows x 128 cols of W_hq.
    if (fullN) {
      for (int rep = 0; rep < 4; ++rep) {
        const int idx = rep * 256 + tid;
        const int nn4 = (idx & 31) * 4;  // 0,4,...,124
        const int kk  = idx >> 5;        // 0..31
        async_g2lds_b128(&Bs[kk][nn4],
                         &Whq[(size_t)(kc + kk) * VOCAB + nBase + nn4]);
      }
    } else {
      for (int rep = 0; rep < 4; ++rep) {
        const int idx = rep * 256 + tid;
        const int nn4 = (idx & 31) * 4;
        const int kk  = idx >> 5;
        for (int e = 0; e < 4; ++e) {
          const int col = nBase + nn4 + e;
          Bs[kk][nn4 + e] =
              (col < VOCAB) ? Whq[(size_t)(kc + kk) * VOCAB + col] : 0.f;
        }
      }
    }
    wait_asynccnt0();   // async DMA landed in LDS
    __syncthreads();    // publish to all waves

    for (int k0 = 0; k0 < TK; k0 += 4) {
      const int ka = k0 + 2 * half;
      v2f afrag[2], bfrag[4];
      for (int a0 = 0; a0 < 2; ++a0) {
        const int row = rowBase + a0 * 16 + i;
        afrag[a0].x = As[row][ka + 0];   // adjacent -> single ds_load_b64
        afrag[a0].y = As[row][ka + 1];
      }
      for (int b0 = 0; b0 < 4; ++b0) {
        bfrag[b0].x = Bs[ka + 0][colBase + b0 * 16 + i];
        bfrag[b0].y = Bs[ka + 1][colBase + b0 * 16 + i];
      }
      for (int a0 = 0; a0 < 2; ++a0)
        for (int b0 = 0; b0 < 4; ++b0)
          acc[a0][b0] = wmma_f32_4(afrag[a0], bfrag[b0], acc[a0][b0]);
    }
    __syncthreads();
  }

  // Epilogue: add bias, store (64-B coalesced segments per half-wave).
  for (int a0 = 0; a0 < 2; ++a0) {
    for (int b0 = 0; b0 < 4; ++b0) {
      const int col = nBase + colBase + b0 * 16 + i;
      if (col < VOCAB) {
        const float bias = bq[col];
        for (int v = 0; v < 8; ++v) {
          const int row = mBase + rowBase + a0 * 16 + half * 8 + v;
          out[(size_t)row * VOCAB + col] = acc[a0][b0][v] + bias;
        }
      }
    }
  }
}

extern "C" void kernel_launch(void* const* d_in, const int* in_sizes, int n_in,
                              void* d_out, int out_size, void* d_ws, size_t ws_size,
                              hipStream_t stream) {
  const int*   X   = (const int*)  d_in[0];
  const float* H0  = (const float*)d_in[1];
  const float* Wxz = (const float*)d_in[2];
  const float* Whz = (const float*)d_in[3];
  const float* bz  = (const float*)d_in[4];
  const float* Wxr = (const float*)d_in[5];
  const float* Whr = (const float*)d_in[6];
  const float* br  = (const float*)d_in[7];
  const float* Wxh = (const float*)d_in[8];
  const float* Whh = (const float*)d_in[9];
  const float* bh  = (const float*)d_in[10];
  const float* Whq = (const float*)d_in[11];
  const float* bq  = (const float*)d_in[12];

  float* out    = (float*)d_out;                            // (8192, 10000)
  float* Hfinal = out + (size_t)STEPS * BATCH * VOCAB;      // tuple tail: (32, 256)
  float* Hall   = (float*)d_ws;                             // (8192, 256), 8 MB scratch

  hipLaunchKernelGGL(gru_recurrence, dim3(1), dim3(1024), 0, stream,
                     X, H0, Wxz, Whz, bz, Wxr, Whr, br, Wxh, Whh, bh,
                     Hall, Hfinal);

  dim3 grid((VOCAB + TN - 1) / TN, (STEPS * BATCH) / TM);   // (79, 64)
  hipLaunchKernelGGL(proj_gemm, grid, dim3(256), 0, stream,
                     Hall, Whq, bq, out);
}